// QRNN_12506944766103
// MI455X (gfx1250) — compile-verified
//
#include <hip/hip_runtime.h>
#include <math.h>

// QRNN complex density-matrix scan on gfx1250 (MI455X).
// h' = U (g h + (1-g) x) U^H per batch, scanned over T steps.
// fp32 WMMA (16x16x4) + register-resident U fragments + TDM double-buffered
// async staging of x into LDS.

typedef float v2f __attribute__((ext_vector_type(2)));
typedef float v8f __attribute__((ext_vector_type(8)));
typedef unsigned int u32x4 __attribute__((ext_vector_type(4)));
typedef int i32x4 __attribute__((ext_vector_type(4)));
typedef int i32x8 __attribute__((ext_vector_type(8)));

#define QD 64    // embed dim
#define QT 128   // time steps
#define QB 64    // batch

__device__ __forceinline__ v8f wmma4(v2f a, v2f b, v8f c) {
  // D = A(16x4 f32) * B(4x16 f32) + C(16x16 f32)
  return __builtin_amdgcn_wmma_f32_16x16x4_f32(
      /*neg_a=*/false, a, /*neg_b=*/false, b,
      /*c_mod=*/(short)0, c, /*reuse_a=*/false, /*reuse_b=*/false);
}

__device__ __forceinline__ v2f lds_ld2(const float* p) {
  v2f r; r.x = p[0]; r.y = p[1]; return r;
}

// TDM: DMA one contiguous 64x64 f32 tile from global into LDS.
// D# built per CDNA5 ISA ch.8 (group0: count/lds/global/type; group1: dims).
// This toolchain exposes the 6-arg builtin: (g0, g1, g2, g3, g4, cpol).
__device__ __forceinline__ void tdm_load_64x64_f32(const float* gsrc,
                                                   unsigned lds_off) {
  const unsigned long long ga = (unsigned long long)(uintptr_t)gsrc;
  u32x4 g0;
  g0.x = 1u;                                        // count=1 (valid user D#)
  g0.y = lds_off;                                   // lds_addr (bytes)
  g0.z = (unsigned)ga;                              // global_addr[31:0]
  g0.w = (unsigned)((ga >> 32) & 0x01FFFFFFu)       // global_addr[56:32]
         | (2u << 30);                              // type=2 ("image")
  i32x8 g1;
  g1[0] = 2 << 16;       // data_size=2 -> 4 bytes; wg_mask=0 (not in cluster)
  g1[1] = QD << 16;      // tensor_dim0 = 64 (low 16 bits, bits 79:48)
  g1[2] = QD << 16;      // tensor_dim1 = 64 (low 16 bits, bits 111:80)
  g1[3] = QD << 16;      // tile_dim0   = 64 (bits 127:112)
  g1[4] = QD;            // tile_dim1   = 64 (bits 143:128), tile_dim2=0
  g1[5] = QD;            // tensor_dim0_stride = 64 (bits 207:160, low 32)
  g1[6] = 0;             // stride high bits / tensor_dim1_stride low
  g1[7] = 0;
  const i32x4 gz4 = {0, 0, 0, 0};                   // 2-D tile: groups 2/3 unused
  const i32x8 gz8 = {0, 0, 0, 0, 0, 0, 0, 0};
  __builtin_amdgcn_tensor_load_to_lds(g0, g1, gz4, gz4, gz8, 0);
}

__global__ __launch_bounds__(512)
void qrnn_scan_kernel(const float* __restrict__ data_r,
                      const float* __restrict__ data_i,
                      const float* __restrict__ U_r,
                      const float* __restrict__ U_i,
                      const float* __restrict__ lam_p,
                      float* __restrict__ out) {
  __shared__ float sUr[QD * QD];
  __shared__ float sUi[QD * QD];
  __shared__ float sHr[QD * QD];       // h, then v (in-place), then h'
  __shared__ float sHi[QD * QD];
  __shared__ float sTr[QD * QD];       // t = U v
  __shared__ float sTi[QD * QD];
  __shared__ float sXr[2][QD * QD];    // TDM double buffer for x_r
  __shared__ float sXi[2][QD * QD];    // TDM double buffer for x_i

  const int tid  = threadIdx.x;
  const int lane = tid & 31;
  const int wave = tid >> 5;            // 0..15 -> 4x4 tile grid
  const int tm   = (wave >> 2) << 4;    // tile row base
  const int tn   = (wave & 3) << 4;     // tile col base
  const int lr   = lane & 15;           // lane row/col within tile
  const int lh2  = (lane >> 4) << 1;    // K sub-offset: 0 or 2 (A/B layout)
  const int lh8  = (lane >> 4) << 3;    // M sub-offset: 0 or 8 (C/D layout)

  const int b = blockIdx.x;

  // ---- stage U into LDS (b128 loads), init h = I / D ----
  {
    const float4* ur4 = (const float4*)U_r;
    const float4* ui4 = (const float4*)U_i;
    float4* dUr = (float4*)sUr;
    float4* dUi = (float4*)sUi;
    for (int i = tid; i < QD * QD / 4; i += 512) {
      dUr[i] = ur4[i];
      dUi[i] = ui4[i];
    }
    for (int i = tid; i < QD * QD; i += 512) {
      const int r = i >> 6, c = i & 63;
      sHr[i] = (r == c) ? (1.0f / (float)QD) : 0.0f;
      sHi[i] = 0.0f;
    }
  }
  const float lam = lam_p[0];
  const float g   = 1.0f / (1.0f + expf(-lam));
  const float gi  = 1.0f - g;

  // ---- TDM preload of x[t=0] into buffer slot 0 (wave 0 issues) ----
  const size_t x0off = (size_t)b * (QD * QD);
  if (wave == 0) {
    tdm_load_64x64_f32(data_r + x0off, (unsigned)(uintptr_t)&sXr[0][0]);
    tdm_load_64x64_f32(data_i + x0off, (unsigned)(uintptr_t)&sXi[0][0]);
    __builtin_amdgcn_s_wait_tensorcnt(0);
  }
  __syncthreads();

  // ---- hoist U fragments into registers (invariant across all T steps) ----
  // Phase-A A-operands: U[tm+lr][k+lh2..+1]; phase-B B-operands: U^T tiles,
  // (U^T)[k'][n] = U[n][k'] -> U[tn+lr][k+lh2..+1] (both contiguous float2).
  v2f fAr[QD / 4], fAi[QD / 4], fBr[QD / 4], fBi[QD / 4];
  #pragma unroll
  for (int kk = 0; kk < QD / 4; ++kk) {
    fAr[kk] = lds_ld2(sUr + (tm + lr) * QD + 4 * kk + lh2);
    fAi[kk] = lds_ld2(sUi + (tm + lr) * QD + 4 * kk + lh2);
    fBr[kk] = lds_ld2(sUr + (tn + lr) * QD + 4 * kk + lh2);
    fBi[kk] = lds_ld2(sUi + (tn + lr) * QD + 4 * kk + lh2);
  }

  for (int t = 0; t < QT; ++t) {
    const size_t xoff = ((size_t)t * QB + b) * (size_t)(QD * QD);
    const int slot = t & 1;

    // ---- mix: v = g*h + (1-g)*x (x from LDS double buffer), in-place ----
    {
      const float4* xr4 = (const float4*)&sXr[slot][0];
      const float4* xi4 = (const float4*)&sXi[slot][0];
      float4* hr4 = (float4*)sHr;
      float4* hi4 = (float4*)sHi;
      #pragma unroll
      for (int i = tid; i < QD * QD / 4; i += 512) {
        float4 xr = xr4[i], xi = xi4[i];
        float4 hr = hr4[i], hi = hi4[i];
        hr.x = g * hr.x + gi * xr.x;  hr.y = g * hr.y + gi * xr.y;
        hr.z = g * hr.z + gi * xr.z;  hr.w = g * hr.w + gi * xr.w;
        hi.x = g * hi.x + gi * xi.x;  hi.y = g * hi.y + gi * xi.y;
        hi.z = g * hi.z + gi * xi.z;  hi.w = g * hi.w + gi * xi.w;
        hr4[i] = hr;
        hi4[i] = hi;
      }
    }
    __syncthreads();

    // ---- async-stage x[t][b+1] (next matrix after xoff) into the other slot while we compute ----
    if (wave == 0 && t + 1 < QT) {
      tdm_load_64x64_f32(data_r + xoff + QD * QD,
                         (unsigned)(uintptr_t)&sXr[slot ^ 1][0]);
      tdm_load_64x64_f32(data_i + xoff + QD * QD,
                         (unsigned)(uintptr_t)&sXi[slot ^ 1][0]);
    }

    // ---- phase A: t = U @ v (complex), tile (tm,tn) per wave ----
    {
      v8f arr = {0,0,0,0,0,0,0,0};
      v8f aii = {0,0,0,0,0,0,0,0};
      v8f ari = {0,0,0,0,0,0,0,0};
      v8f air = {0,0,0,0,0,0,0,0};
      #pragma unroll
      for (int kk = 0; kk < QD / 4; ++kk) {
        const int k = 4 * kk;
        // B tiles of v (4x16): lane holds v[k+lh2..+1][tn+lr] (column pair)
        v2f Br, Bi;
        Br.x = sHr[(k + lh2 + 0) * QD + tn + lr];
        Br.y = sHr[(k + lh2 + 1) * QD + tn + lr];
        Bi.x = sHi[(k + lh2 + 0) * QD + tn + lr];
        Bi.y = sHi[(k + lh2 + 1) * QD + tn + lr];
        arr = wmma4(fAr[kk], Br, arr);   // U_r v_r
        aii = wmma4(fAi[kk], Bi, aii);   // U_i v_i
        ari = wmma4(fAr[kk], Bi, ari);   // U_r v_i
        air = wmma4(fAi[kk], Br, air);   // U_i v_r
      }
      // t_r = U_r v_r - U_i v_i ; t_i = U_r v_i + U_i v_r
      #pragma unroll
      for (int i = 0; i < 8; ++i) {
        const int r = tm + i + lh8;
        const int c = tn + lr;
        sTr[r * QD + c] = arr[i] - aii[i];
        sTi[r * QD + c] = ari[i] + air[i];
      }
    }
    __syncthreads();

    // ---- phase B: h' = t @ U^H ; U^H = U_r^T - i U_i^T ----
    {
      v8f brr = {0,0,0,0,0,0,0,0};
      v8f bii = {0,0,0,0,0,0,0,0};
      v8f bri = {0,0,0,0,0,0,0,0};
      v8f bir = {0,0,0,0,0,0,0,0};
      #pragma unroll
      for (int kk = 0; kk < QD / 4; ++kk) {
        const int k = 4 * kk;
        // A tiles of t (16x4): contiguous float2 per lane
        v2f Atr = lds_ld2(sTr + (tm + lr) * QD + k + lh2);
        v2f Ati = lds_ld2(sTi + (tm + lr) * QD + k + lh2);
        brr = wmma4(Atr, fBr[kk], brr);  // t_r U_r^T
        bii = wmma4(Ati, fBi[kk], bii);  // t_i U_i^T
        bri = wmma4(Ati, fBr[kk], bri);  // t_i U_r^T
        bir = wmma4(Atr, fBi[kk], bir);  // t_r U_i^T
      }
      // h'_r = t_r U_r^T + t_i U_i^T ; h'_i = t_i U_r^T - t_r U_i^T
      float* outR = out + xoff;                                  // [0,t,b]
      float* outI = out + (size_t)QT * QB * QD * QD + xoff;      // [1,t,b]
      #pragma unroll
      for (int i = 0; i < 8; ++i) {
        const int r = tm + i + lh8;
        const int c = tn + lr;
        const float hr = brr[i] + bii[i];
        const float hi = bri[i] - bir[i];
        sHr[r * QD + c] = hr;
        sHi[r * QD + c] = hi;
        outR[r * QD + c] = hr;
        outI[r * QD + c] = hi;
      }
    }
    // wave 0 guarantees next-step x is resident before releasing the barrier
    if (wave == 0) {
      __builtin_amdgcn_s_wait_tensorcnt(0);
    }
    __syncthreads();
  }
}

extern "C" void kernel_launch(void* const* d_in, const int* in_sizes, int n_in,
                              void* d_out, int out_size, void* d_ws, size_t ws_size,
                              hipStream_t stream) {
  (void)in_sizes; (void)n_in; (void)out_size; (void)d_ws; (void)ws_size;
  const float* data_r = (const float*)d_in[0];
  const float* data_i = (const float*)d_in[1];
  const float* U_r    = (const float*)d_in[2];
  const float* U_i    = (const float*)d_in[3];
  const float* lam    = (const float*)d_in[4];
  float* out = (float*)d_out;

  qrnn_scan_kernel<<<dim3(QB), dim3(512), 0, stream>>>(
      data_r, data_i, U_r, U_i, lam, out);
}